// Graph_atten_layer_45449343926931
// MI455X (gfx1250) — compile-verified
//
#include <hip/hip_runtime.h>

typedef float v2f __attribute__((ext_vector_type(2)));
typedef float v4f __attribute__((ext_vector_type(4)));
typedef float v8f __attribute__((ext_vector_type(8)));
typedef int   v4i __attribute__((ext_vector_type(4)));

#define N_NODES 8192
#define DIN     256
#define DOUT    64
#define SLOPE   0.01f
#define CHUNK   1024
#define PSTRIDE 1028  // 1024 + 4 pad: A-operand LDS reads hit distinct banks (1028%64==4)

__device__ __forceinline__ float lrelu(float x) { return fmaxf(x, SLOPE * x); }

// ---------------------------------------------------------------------------
// Kernel 1: wh = inputs @ w  (8192x256 @ 256x64) with V_WMMA_F32_16X16X4_F32.
// One wave computes a 16x64 tile (4 C tiles), K-loop of 64 steps of 4.
// ---------------------------------------------------------------------------
__global__ void k_wh(const float* __restrict__ inp, const float* __restrict__ wmat,
                     float* __restrict__ wh) {
  const int tid  = threadIdx.x;
  const int wave = tid >> 5;
  const int lane = tid & 31;
  const int half = lane >> 4;   // K sub-half per ISA A/B layout
  const int m16  = lane & 15;
  const int i0   = (blockIdx.x * 8 + wave) * 16;

  v8f c0 = {}, c1 = {}, c2 = {}, c3 = {};
  const float* arow = inp + (size_t)(i0 + m16) * DIN + 2 * half;

  for (int k0 = 0; k0 < DIN; k0 += 4) {
    v2f a;
    a.x = arow[k0];
    a.y = arow[k0 + 1];
    const float* bb = wmat + (size_t)(k0 + 2 * half) * DOUT + m16;
    v2f b;
    b.x = bb[0];            b.y = bb[DOUT];
    c0 = __builtin_amdgcn_wmma_f32_16x16x4_f32(false, a, false, b, (short)0, c0, false, false);
    b.x = bb[16];           b.y = bb[DOUT + 16];
    c1 = __builtin_amdgcn_wmma_f32_16x16x4_f32(false, a, false, b, (short)0, c1, false, false);
    b.x = bb[32];           b.y = bb[DOUT + 32];
    c2 = __builtin_amdgcn_wmma_f32_16x16x4_f32(false, a, false, b, (short)0, c2, false, false);
    b.x = bb[48];           b.y = bb[DOUT + 48];
    c3 = __builtin_amdgcn_wmma_f32_16x16x4_f32(false, a, false, b, (short)0, c3, false, false);
  }

#pragma unroll
  for (int v = 0; v < 8; ++v) {
    const size_t row = (size_t)(i0 + v + 8 * half);
    wh[row * DOUT +  0 + m16] = c0[v];
    wh[row * DOUT + 16 + m16] = c1[v];
    wh[row * DOUT + 32 + m16] = c2[v];
    wh[row * DOUT + 48 + m16] = c3[v];
  }
}

// ---------------------------------------------------------------------------
// Kernel 2: s_src[i] = wh[i,:].a[:64], s_dst[i] = wh[i,:].a[64:]
// ---------------------------------------------------------------------------
__global__ void k_s(const float* __restrict__ wh, const float* __restrict__ avec,
                    float* __restrict__ s_src, float* __restrict__ s_dst) {
  __shared__ float av[2 * DOUT];
  const int tid = threadIdx.x;
  if (tid < 2 * DOUT) av[tid] = avec[tid];
  __syncthreads();
  const int row = blockIdx.x * blockDim.x + tid;
  const float* wr = wh + (size_t)row * DOUT;
  float ss = 0.f, sd = 0.f;
#pragma unroll 8
  for (int d = 0; d < DOUT; ++d) {
    const float v = wr[d];
    ss = fmaf(v, av[d], ss);
    sd = fmaf(v, av[DOUT + d], sd);
  }
  s_src[row] = ss;
  s_dst[row] = sd;
}

// ---------------------------------------------------------------------------
// Kernel 3: fused masked-softmax + (att @ wh) + lrelu.
// One block = 16 rows, 256 threads (8 waves). Single pass over adj (268 MB,
// the HBM-bound stream): b128 non-temporal loads (adj is touched once and
// exceeds L2; keep wh/s_dst resident). m_i = lrelu(s_src[i] + max(s_dst)) is
// a valid softmax shift (lrelu monotone) => exp never overflows, no rescale.
// exp-score chunk (16x1024) staged in LDS as the WMMA A operand; fp32
// 16x16x4 WMMA accumulates p @ wh. Waves 0-3: N-tiles 0-3 with K-half 0;
// waves 4-7: same tiles, K-half 1; partials combined through LDS.
// ---------------------------------------------------------------------------
__global__ void __launch_bounds__(256) k_att(const int* __restrict__ adj,
                                             const float* __restrict__ wh,
                                             const float* __restrict__ s_src,
                                             const float* __restrict__ s_dst,
                                             float* __restrict__ out) {
  __shared__ float sdst_s[N_NODES];           // 32 KB, reused all chunks
  __shared__ float p_s[16 * PSTRIDE];         // 64.25 KB exp-score chunk (A operand)
  __shared__ float ssrc_s[16];
  __shared__ float m_s[16];
  __shared__ float l_s[16];
  __shared__ float linv_s[16];
  __shared__ float red_s[256];
  __shared__ float cx_s[4 * 256];             // C-tile transfer waves 4-7 -> 0-3

  const int tid  = threadIdx.x;
  const int wave = tid >> 5;
  const int lane = tid & 31;
  const int half = lane >> 4;
  const int m16  = lane & 15;
  const int i0   = blockIdx.x * 16;
  const int n_tile = wave & 3;
  const int khalf  = wave >> 2;

  // Stage s_dst into LDS + block-wide max(s_dst)
  float lmax = -3.0e38f;
  for (int c = tid; c < N_NODES; c += 256) {
    const float v = s_dst[c];
    sdst_s[c] = v;
    lmax = fmaxf(lmax, v);
  }
  red_s[tid] = lmax;
  if (tid < 16) { ssrc_s[tid] = s_src[i0 + tid]; l_s[tid] = 0.f; }
  __syncthreads();
  for (int s = 128; s > 0; s >>= 1) {
    if (tid < s) red_s[tid] = fmaxf(red_s[tid], red_s[tid + s]);
    __syncthreads();
  }
  if (tid < 16) m_s[tid] = lrelu(ssrc_s[tid] + red_s[0]);  // >= all row-i scores
  __syncthreads();

  float ssr[16], mm[16], l16[16];
#pragma unroll
  for (int r = 0; r < 16; ++r) { ssr[r] = ssrc_s[r]; mm[r] = m_s[r]; l16[r] = 0.f; }

  v8f acc = {};

  for (int c0 = 0; c0 < N_NODES; c0 += CHUNK) {
    // --- score phase: each thread owns 4 consecutive columns (b128 loads) ---
    const v4f sd4 = *(const v4f*)&sdst_s[c0 + 4 * tid];
    const v4i* abase = (const v4i*)(adj + (size_t)i0 * N_NODES + c0) + tid;
#pragma unroll
    for (int r = 0; r < 16; ++r) {
      const v4i a4 = __builtin_nontemporal_load(abase + (size_t)r * (N_NODES / 4));
      v4f p4;
      p4.x = (a4.x > 0) ? __expf(lrelu(ssr[r] + sd4.x) - mm[r]) : 0.0f;
      p4.y = (a4.y > 0) ? __expf(lrelu(ssr[r] + sd4.y) - mm[r]) : 0.0f;
      p4.z = (a4.z > 0) ? __expf(lrelu(ssr[r] + sd4.z) - mm[r]) : 0.0f;
      p4.w = (a4.w > 0) ? __expf(lrelu(ssr[r] + sd4.w) - mm[r]) : 0.0f;
      *(v4f*)&p_s[r * PSTRIDE + 4 * tid] = p4;   // ds_store_b128, conflict-free
      l16[r] += (p4.x + p4.y) + (p4.z + p4.w);
    }
    __syncthreads();

    // --- WMMA phase: acc += p_chunk(16xK) @ wh(Kx16 tile), K-half per wave ---
    const float* pa = p_s + m16 * PSTRIDE + khalf * (CHUNK / 2) + 2 * half;
    const float* bb = wh + (size_t)(c0 + khalf * (CHUNK / 2) + 2 * half) * DOUT
                         + n_tile * 16 + m16;
#pragma unroll 4
    for (int kk = 0; kk < CHUNK / 2; kk += 4) {
      v2f a;
      a.x = pa[kk];
      a.y = pa[kk + 1];
      v2f b;
      b.x = bb[(size_t)kk * DOUT];
      b.y = bb[(size_t)(kk + 1) * DOUT];
      acc = __builtin_amdgcn_wmma_f32_16x16x4_f32(false, a, false, b, (short)0, acc, false, false);
    }
    __syncthreads();
  }

  // --- finalize: reduce l, combine K-half partials, normalize + lrelu ---
#pragma unroll
  for (int r = 0; r < 16; ++r) atomicAdd(&l_s[r], l16[r]);
  __syncthreads();
  if (tid < 16) linv_s[tid] = 1.0f / l_s[tid];
  if (wave >= 4) {
#pragma unroll
    for (int v = 0; v < 8; ++v) cx_s[(wave - 4) * 256 + v * 32 + lane] = acc[v];
  }
  __syncthreads();
  if (wave < 4) {
#pragma unroll
    for (int v = 0; v < 8; ++v) {
      const int row = v + 8 * half;
      const float val = (acc[v] + cx_s[wave * 256 + v * 32 + lane]) * linv_s[row];
      __builtin_nontemporal_store(lrelu(val),
          &out[(size_t)(i0 + row) * DOUT + n_tile * 16 + m16]);
    }
  }
}

// ---------------------------------------------------------------------------
extern "C" void kernel_launch(void* const* d_in, const int* in_sizes, int n_in,
                              void* d_out, int out_size, void* d_ws, size_t ws_size,
                              hipStream_t stream) {
  const float* inp  = (const float*)d_in[0];  // [8192,256]
  const int*   adj  = (const int*)  d_in[1];  // [8192,8192]
  const float* wmat = (const float*)d_in[2];  // [256,64]
  const float* avec = (const float*)d_in[3];  // [128,1]
  float* out = (float*)d_out;                 // [8192,64]

  float* wh    = (float*)d_ws;                // 8192*64
  float* s_src = wh + (size_t)N_NODES * DOUT; // 8192
  float* s_dst = s_src + N_NODES;             // 8192

  k_wh <<<N_NODES / (8 * 16), 256, 0, stream>>>(inp, wmat, wh);
  k_s  <<<N_NODES / 256,      256, 0, stream>>>(wh, avec, s_src, s_dst);
  k_att<<<N_NODES / 16,       256, 0, stream>>>(adj, wh, s_src, s_dst, out);
}